// FasterRCNN_31860067402141
// MI455X (gfx1250) — compile-verified
//
#include <hip/hip_runtime.h>
#include <math.h>

#define N_CLS   80
#define N_CP1   81
#define N_ROI   2000
#define N_B     16
#define TOPK    300
#define SORT_N  2048
#define SCORE_THR 0.01f
#define NMS_THR   0.3f
#define NEG_BIG  -3.4e38f

typedef int v4i __attribute__((ext_vector_type(4)));
typedef __attribute__((address_space(1))) v4i gm_v4i_t;   // global
typedef __attribute__((address_space(3))) v4i lds_v4i_t;  // LDS

// -------- CDNA5 async global->LDS copy (ASYNCcnt path) --------
__device__ __forceinline__ void async_b128_to_lds(const void* g, void* l) {
#if __has_builtin(__builtin_amdgcn_global_load_async_to_lds_b128)
  __builtin_amdgcn_global_load_async_to_lds_b128(
      (gm_v4i_t*)(uintptr_t)g,
      (lds_v4i_t*)(unsigned)(uintptr_t)l, 0, 0);
#else
  unsigned lofs = (unsigned)(uintptr_t)l;   // low 32b of flat LDS ptr = LDS offset
  asm volatile("global_load_async_to_lds_b128 %0, %1, off"
               :: "v"(lofs), "v"(g) : "memory");
#endif
}

__device__ __forceinline__ void wait_async0() {
#if __has_builtin(__builtin_amdgcn_s_wait_asynccnt)
  __builtin_amdgcn_s_wait_asynccnt(0);
#else
  asm volatile("s_wait_asynccnt 0x0" ::: "memory");
#endif
}

// ============ Kernel 1: softmax(81) + threshold mask, one wave per ROI ============
__global__ __launch_bounds__(256) void k_softmax_mask(
    const float* __restrict__ score,   // [B][R][81]
    float* __restrict__ masked)        // [B*80][R]
{
  const int lane = threadIdx.x & 31;
  const int wid  = threadIdx.x >> 5;
  const int row  = blockIdx.x * 8 + wid;           // 0..31999
  if (row >= N_B * N_ROI) return;
  const int b = row / N_ROI;
  const int r = row - b * N_ROI;
  const float* s = score + (size_t)row * N_CP1;

  const int c0 = lane, c1 = lane + 32, c2 = lane + 64;
  float v0 = s[c0];
  float v1 = (c1 < N_CP1) ? s[c1] : NEG_BIG;
  float v2 = (c2 < N_CP1) ? s[c2] : NEG_BIG;

  float m = fmaxf(v0, fmaxf(v1, v2));
  #pragma unroll
  for (int o = 16; o > 0; o >>= 1) m = fmaxf(m, __shfl_xor(m, o, 32));

  float e0 = expf(v0 - m);
  float e1 = (c1 < N_CP1) ? expf(v1 - m) : 0.f;
  float e2 = (c2 < N_CP1) ? expf(v2 - m) : 0.f;
  float ssum = e0 + e1 + e2;
  #pragma unroll
  for (int o = 16; o > 0; o >>= 1) ssum += __shfl_xor(ssum, o, 32);
  const float inv = 1.0f / ssum;

  // foreground classes 1..80 -> slot (c-1); layout [b*80 + cc][r]
  if (c0 >= 1) {
    float p = e0 * inv;
    masked[((size_t)(b * N_CLS + (c0 - 1))) * N_ROI + r] = (p > SCORE_THR) ? p : -1.0f;
  }
  {
    float p = e1 * inv;
    masked[((size_t)(b * N_CLS + (c1 - 1))) * N_ROI + r] = (p > SCORE_THR) ? p : -1.0f;
  }
  if (c2 < N_CP1) {
    float p = e2 * inv;
    masked[((size_t)(b * N_CLS + (c2 - 1))) * N_ROI + r] = (p > SCORE_THR) ? p : -1.0f;
  }
}

// ============ Kernel 2: per (b,c) bitonic top-300 (stable ties) + lazy box decode ============
__global__ __launch_bounds__(256) void k_topk_decode(
    const float* __restrict__ masked,  // [P][R]
    const float* __restrict__ rois,    // [B][R][4]
    const float* __restrict__ loc,     // [B][R][324]
    const int*   __restrict__ imgh_p,
    const int*   __restrict__ imgw_p,
    float* __restrict__ tboxes,        // [P][300][4]
    float* __restrict__ tscore,        // [P][300]
    int*   __restrict__ tvalid)        // [P][300]
{
  __shared__ float skey[SORT_N];
  __shared__ int   sidx[SORT_N];
  const int p   = blockIdx.x;              // 0..1279
  const int b   = p / N_CLS;
  const int cc  = p - b * N_CLS;           // fg slot; true class = cc+1
  const int tid = threadIdx.x;

  for (int i = tid; i < SORT_N; i += 256) {
    if (i < N_ROI) { skey[i] = masked[(size_t)p * N_ROI + i]; sidx[i] = i; }
    else           { skey[i] = NEG_BIG;                        sidx[i] = 0x7fffffff; }
  }
  __syncthreads();

  // Bitonic sort, descending by key, ascending index on ties (== jax top_k order)
  for (int k = 2; k <= SORT_N; k <<= 1) {
    for (int j = k >> 1; j > 0; j >>= 1) {
      for (int q = tid; q < SORT_N / 2; q += 256) {
        int i   = ((q & ~(j - 1)) << 1) | (q & (j - 1));
        int ixj = i + j;
        float ka = skey[i], kb = skey[ixj];
        int   ia = sidx[i], ib = sidx[ixj];
        bool a_less_b = (ka < kb) || (ka == kb && ia > ib);  // strict total order
        bool desc = ((i & k) == 0);
        bool dosw = desc ? a_less_b : !a_less_b;
        if (dosw) { skey[i] = kb; skey[ixj] = ka; sidx[i] = ib; sidx[ixj] = ia; }
      }
      __syncthreads();
    }
  }

  const float H = (float)(*imgh_p);
  const float W = (float)(*imgw_p);
  for (int kk = tid; kk < TOPK; kk += 256) {
    float s = skey[kk];
    int   r = sidx[kk];
    if (r < 0 || r >= N_ROI) r = 0;
    const float* rp = rois + ((size_t)b * N_ROI + r) * 4;
    const float* lp = loc  + ((size_t)b * N_ROI + r) * (size_t)(N_CP1 * 4)
                           + (size_t)(cc + 1) * 4;
    float y0 = rp[0], x0 = rp[1], y1 = rp[2], x1 = rp[3];
    float h  = y1 - y0,  w  = x1 - x0;
    float cy = y0 + 0.5f * h, cx = x0 + 0.5f * w;
    float dy = lp[0], dx = lp[1], dh = lp[2], dw = lp[3];
    float ncy = dy * h + cy, ncx = dx * w + cx;
    float nh  = expf(dh) * h, nw = expf(dw) * w;
    float by0 = fminf(fmaxf(ncy - 0.5f * nh, 0.f), H);
    float bx0 = fminf(fmaxf(ncx - 0.5f * nw, 0.f), W);
    float by1 = fminf(fmaxf(ncy + 0.5f * nh, 0.f), H);
    float bx1 = fminf(fmaxf(ncx + 0.5f * nw, 0.f), W);
    float* op = tboxes + ((size_t)p * TOPK + kk) * 4;
    op[0] = by0; op[1] = bx0; op[2] = by1; op[3] = bx1;
    tscore[(size_t)p * TOPK + kk] = s;
    tvalid[(size_t)p * TOPK + kk] = (s > SCORE_THR) ? 1 : 0;
  }
}

// ============ Kernel 3: NMS (async-LDS staged tile) + final output ============
__global__ __launch_bounds__(320) void k_nms_out(
    const float* __restrict__ tboxes,  // [P][300][4], 16B aligned
    const float* __restrict__ tscore,  // [P][300]
    const int*   __restrict__ tvalid,  // [P][300]
    float* __restrict__ out)           // [P][300][6]
{
  __shared__ __align__(16) float lbox[TOPK * 4];
  __shared__ float larea[TOPK];
  __shared__ int   lvalid[TOPK];
  __shared__ int   lsupp[TOPK];
  const int p = blockIdx.x;
  const int t = threadIdx.x;

  if (t < TOPK) {
    // async tile fetch: 300 lanes x 16B = 4.8KB, overlapped with valid-flag loads
    async_b128_to_lds(tboxes + ((size_t)p * TOPK + t) * 4, &lbox[t * 4]);
    lvalid[t] = tvalid[(size_t)p * TOPK + t];
    lsupp[t]  = 0;
  }
  wait_async0();
  __syncthreads();

  if (t < TOPK) {
    float y0 = lbox[t * 4], x0 = lbox[t * 4 + 1], y1 = lbox[t * 4 + 2], x1 = lbox[t * 4 + 3];
    larea[t] = fmaxf(y1 - y0, 0.f) * fmaxf(x1 - x0, 0.f);
  }

  for (int i = 0; i < TOPK; ++i) {
    __syncthreads();
    bool keep_i = (lvalid[i] != 0) && (lsupp[i] == 0);
    if (keep_i && t > i && t < TOPK) {
      float iy0 = lbox[i * 4], ix0 = lbox[i * 4 + 1], iy1 = lbox[i * 4 + 2], ix1 = lbox[i * 4 + 3];
      float ty0 = lbox[t * 4], tx0 = lbox[t * 4 + 1], ty1 = lbox[t * 4 + 2], tx1 = lbox[t * 4 + 3];
      float ih = fmaxf(fminf(iy1, ty1) - fmaxf(iy0, ty0), 0.f);
      float iw = fmaxf(fminf(ix1, tx1) - fmaxf(ix0, tx0), 0.f);
      float inter = ih * iw;
      float iou = inter / (larea[i] + larea[t] - inter + 1e-9f);
      if (iou > NMS_THR) lsupp[t] = 1;
    }
  }
  __syncthreads();

  if (t < TOPK) {
    bool keep = (lvalid[t] != 0) && (lsupp[t] == 0);
    float s = tscore[(size_t)p * TOPK + t];
    float* op = out + ((size_t)p * TOPK + t) * 6;
    op[0] = lbox[t * 4];     op[1] = lbox[t * 4 + 1];
    op[2] = lbox[t * 4 + 2]; op[3] = lbox[t * 4 + 3];
    op[4] = keep ? s : 0.0f;
    op[5] = keep ? 1.0f : 0.0f;
  }
}

extern "C" void kernel_launch(void* const* d_in, const int* in_sizes, int n_in,
                              void* d_out, int out_size, void* d_ws, size_t ws_size,
                              hipStream_t stream) {
  const float* rois  = (const float*)d_in[0];   // 16*2000*4
  const float* loc   = (const float*)d_in[1];   // 16*2000*324
  const float* score = (const float*)d_in[2];   // 16*2000*81
  const int*   imgh  = (const int*)d_in[3];
  const int*   imgw  = (const int*)d_in[4];
  float* out = (float*)d_out;                   // 16*80*300*6

  const size_t P = (size_t)N_B * N_CLS;         // 1280
  float* ws_masked = (float*)d_ws;                          // P*2000
  float* ws_boxes  = ws_masked + P * N_ROI;                 // P*300*4 (16B aligned)
  float* ws_tscore = ws_boxes + P * TOPK * 4;               // P*300
  int*   ws_valid  = (int*)(ws_tscore + P * TOPK);          // P*300

  (void)in_sizes; (void)n_in; (void)out_size; (void)ws_size;

  // Kernel 1: 32000 ROI rows, one wave per row, 8 rows/block
  k_softmax_mask<<<(N_B * N_ROI) / 8, 256, 0, stream>>>(score, ws_masked);

  // Kernel 2: one block per (b, class)
  k_topk_decode<<<(int)P, 256, 0, stream>>>(ws_masked, rois, loc, imgh, imgw,
                                            ws_boxes, ws_tscore, ws_valid);

  // Kernel 3: one block per (b, class), 10 waves
  k_nms_out<<<(int)P, 320, 0, stream>>>(ws_boxes, ws_tscore, ws_valid, out);
}